// SensorGAT_20950850470230
// MI455X (gfx1250) — compile-verified
//
#include <hip/hip_runtime.h>
#include <math.h>

#define NNODES 50000
#define FIN    64
#define HIDC   32
#define NHEADS 4
#define NCLS   3

typedef __attribute__((ext_vector_type(2))) float v2f;
typedef __attribute__((ext_vector_type(8))) float v8f;

// ---------------- WMMA f32 GEMM: C[M,Nc] = A[M,K] @ B[K,Nc], row-major ----------------
// One wave32 per 16x16 output tile; K stepped by 4 via V_WMMA_F32_16X16X4_F32.
// f32 A 16x4 layout: lane L holds row M=L%16; lanes 0-15 -> K={0,1}, lanes 16-31 -> K={2,3}.
// f32 B 4x16 layout: lane L holds col N=L%16; same K split across lane halves.
// f32 D 16x16 layout: VGPR i holds M = i + 8*(lane/16), N = lane%16.
__global__ void wmma_gemm_f32(const float* __restrict__ A, const float* __restrict__ B,
                              float* __restrict__ C, int M, int K, int Nc) {
  int wave = (blockIdx.x * blockDim.x + threadIdx.x) >> 5;
  int lane = threadIdx.x & 31;
  int tilesN = Nc >> 4;
  int tilesM = M >> 4;
  if (wave >= tilesM * tilesN) return;
  int tm = wave / tilesN;
  int tn = wave - tm * tilesN;
  int r      = lane & 15;
  int khalf  = (lane >> 4) << 1;          // 0 or 2
  const float* Arow = A + (size_t)(tm * 16 + r) * K;
  const float* Bcol = B + tn * 16 + r;
  v8f acc = {};
  #pragma unroll 4
  for (int k = 0; k < K; k += 4) {
    v2f a, b;
    a.x = Arow[k + khalf];
    a.y = Arow[k + khalf + 1];
    b.x = Bcol[(size_t)(k + khalf) * Nc];
    b.y = Bcol[(size_t)(k + khalf + 1) * Nc];
    acc = __builtin_amdgcn_wmma_f32_16x16x4_f32(false, a, false, b, (short)0, acc,
                                                false, false);
  }
  int rowBase = tm * 16 + ((lane >> 4) << 3);
  int col     = tn * 16 + r;
  union { v8f v; float f[8]; } u; u.v = acc;
  #pragma unroll
  for (int i = 0; i < 8; ++i)
    C[(size_t)(rowBase + i) * Nc + col] = u.f[i];
}

// ---------------- helpers ----------------
__global__ void fill_f32(float* p, float v, long n) {
  long i = (long)blockIdx.x * blockDim.x + threadIdx.x;
  if (i < n) p[i] = v;
}

// sign-aware float atomic max using integer atomics
__device__ __forceinline__ void atomicMaxF(float* addr, float val) {
  if (val >= 0.0f)
    atomicMax((int*)addr, __float_as_int(val));
  else
    atomicMin((unsigned int*)addr, __float_as_uint(val));
}

__device__ __forceinline__ void edge_ends(const int* __restrict__ src,
                                          const int* __restrict__ dst,
                                          int e, int E, int& s, int& d) {
  if (e < E) { s = src[e]; d = dst[e]; }
  else       { s = e - E; d = e - E; }       // self-loops appended after real edges
}

// a_src[n,h] = <h[n,h,:], att_src[h,:]>, a_dst likewise
__global__ void attn_coef(const float* __restrict__ h, const float* __restrict__ att_src,
                          const float* __restrict__ att_dst,
                          float* __restrict__ asrc, float* __restrict__ adst,
                          int n, int heads, int C) {
  long i = (long)blockIdx.x * blockDim.x + threadIdx.x;
  if (i >= (long)n * heads) return;
  int node = (int)(i / heads);
  int hd   = (int)(i - (long)node * heads);
  const float* hp = h + (size_t)node * heads * C + (size_t)hd * C;
  const float* as = att_src + (size_t)hd * C;
  const float* ad = att_dst + (size_t)hd * C;
  float s1 = 0.f, s2 = 0.f;
  for (int c = 0; c < C; ++c) { s1 += hp[c] * as[c]; s2 += hp[c] * ad[c]; }
  asrc[i] = s1;
  adst[i] = s2;
}

// pass 1: e = leaky_relu(a_src[src]+a_dst[dst]); segment max into mbuf[dst]
__global__ void edge_score_max(const int* __restrict__ src, const int* __restrict__ dst,
                               int E, int n, int heads,
                               const float* __restrict__ asrc, const float* __restrict__ adst,
                               float* __restrict__ escore, float* __restrict__ mbuf) {
  long i = (long)blockIdx.x * blockDim.x + threadIdx.x;
  long total = (long)(E + n) * heads;
  if (i >= total) return;
  int e  = (int)(i / heads);
  int hd = (int)(i - (long)e * heads);
  int s, d; edge_ends(src, dst, e, E, s, d);
  float v = asrc[(long)s * heads + hd] + adst[(long)d * heads + hd];
  v = (v > 0.f) ? v : 0.2f * v;
  escore[i] = v;
  atomicMaxF(&mbuf[(long)d * heads + hd], v);
}

// pass 2: p = exp(e - m[dst]); segment sum into den[dst]
__global__ void edge_exp_sum(const int* __restrict__ src, const int* __restrict__ dst,
                             int E, int n, int heads,
                             float* __restrict__ escore, const float* __restrict__ mbuf,
                             float* __restrict__ den) {
  long i = (long)blockIdx.x * blockDim.x + threadIdx.x;
  long total = (long)(E + n) * heads;
  if (i >= total) return;
  int e  = (int)(i / heads);
  int hd = (int)(i - (long)e * heads);
  int s, d; edge_ends(src, dst, e, E, s, d);
  float p = expf(escore[i] - mbuf[(long)d * heads + hd]);
  escore[i] = p;
  atomicAdd(&den[(long)d * heads + hd], p);
}

// pass 3: alpha = p / den[dst]
__global__ void edge_norm(const int* __restrict__ src, const int* __restrict__ dst,
                          int E, int n, int heads,
                          float* __restrict__ escore, const float* __restrict__ den) {
  long i = (long)blockIdx.x * blockDim.x + threadIdx.x;
  long total = (long)(E + n) * heads;
  if (i >= total) return;
  int e  = (int)(i / heads);
  int hd = (int)(i - (long)e * heads);
  int s, d; edge_ends(src, dst, e, E, s, d);
  escore[i] = escore[i] / den[(long)d * heads + hd];
}

// pass 4: out[dst, f] += h[src, f] * alpha[e, head(f)]   (thread per (edge, feature))
__global__ void edge_agg(const int* __restrict__ src, const int* __restrict__ dst,
                         int E, int n, int heads, int C,
                         const float* __restrict__ h, const float* __restrict__ alpha,
                         float* __restrict__ out) {
  long i = (long)blockIdx.x * blockDim.x + threadIdx.x;
  long HC = (long)heads * C;
  long total = (long)(E + n) * HC;
  if (i >= total) return;
  int e = (int)(i / HC);
  int f = (int)(i - (long)e * HC);
  int hd = f / C;
  int s, d; edge_ends(src, dst, e, E, s, d);
  float a = alpha[(long)e * heads + hd];
  atomicAdd(&out[(long)d * HC + f], h[(long)s * HC + f] * a);
}

__global__ void bias_elu(float* __restrict__ h, const float* __restrict__ b, int n, int HC) {
  long i = (long)blockIdx.x * blockDim.x + threadIdx.x;
  if (i >= (long)n * HC) return;
  float v = h[i] + b[i % HC];
  h[i] = (v > 0.f) ? v : (expf(v) - 1.f);
}

// column sums of out2[N,32] -> g[32] (LDS partials, 32 atomics per block)
__global__ void colsum32(const float* __restrict__ a, float* __restrict__ g, int n) {
  __shared__ float sm[256];
  int c = threadIdx.x & 31;
  int rlane = threadIdx.x >> 5;             // 0..7
  int rowsPerBlk = blockDim.x >> 5;
  float s = 0.f;
  for (int r = blockIdx.x * rowsPerBlk + rlane; r < n; r += gridDim.x * rowsPerBlk)
    s += a[(size_t)r * 32 + c];
  sm[threadIdx.x] = s;
  __syncthreads();
  if (threadIdx.x < 32) {
    float t = 0.f;
    #pragma unroll
    for (int j = 0; j < 8; ++j) t += sm[threadIdx.x + 32 * j];
    atomicAdd(&g[threadIdx.x], t);
  }
}

// gm = g/N + b2 ; logits = gm @ linW + linb ; softmax -> out[3]
__global__ void finalize(const float* __restrict__ g, const float* __restrict__ b2,
                         const float* __restrict__ linW, const float* __restrict__ linb,
                         float* __restrict__ out, int n) {
  if (threadIdx.x != 0 || blockIdx.x != 0) return;
  float gm[HIDC];
  for (int c = 0; c < HIDC; ++c) gm[c] = g[c] / (float)n + b2[c];
  float lg[NCLS];
  for (int j = 0; j < NCLS; ++j) {
    float acc = linb[j];
    for (int c = 0; c < HIDC; ++c) acc += gm[c] * linW[c * NCLS + j];
    lg[j] = acc;
  }
  float mx = lg[0];
  for (int j = 1; j < NCLS; ++j) mx = lg[j] > mx ? lg[j] : mx;
  float s = 0.f;
  for (int j = 0; j < NCLS; ++j) { lg[j] = expf(lg[j] - mx); s += lg[j]; }
  for (int j = 0; j < NCLS; ++j) out[j] = lg[j] / s;
}

static inline long cdivl(long a, long b) { return (a + b - 1) / b; }

extern "C" void kernel_launch(void* const* d_in, const int* in_sizes, int n_in,
                              void* d_out, int out_size, void* d_ws, size_t ws_size,
                              hipStream_t stream) {
  const float* x        = (const float*)d_in[0];
  const float* W1       = (const float*)d_in[1];
  const float* att_src1 = (const float*)d_in[2];
  const float* att_dst1 = (const float*)d_in[3];
  const float* b1       = (const float*)d_in[4];
  const float* W2       = (const float*)d_in[5];
  const float* att_src2 = (const float*)d_in[6];
  const float* att_dst2 = (const float*)d_in[7];
  const float* b2       = (const float*)d_in[8];
  const float* linW     = (const float*)d_in[9];
  const float* linb     = (const float*)d_in[10];
  const int*   ei       = (const int*)d_in[11];

  const int N  = NNODES;
  const int E  = in_sizes[11] / 2;
  const int EA = E + N;
  const int* srcI = ei;
  const int* dstI = ei + E;

  // ---- workspace layout (floats) ----
  float* ws = (float*)d_ws;
  size_t o = 0;
  float* h1    = ws + o; o += (size_t)N * 128;     // layer-1 features [N,4,32]
  float* out1  = ws + o; o += (size_t)N * 128;     // layer-1 aggregated output
  float* alpha = ws + o; o += (size_t)EA * 4;      // edge scores/alphas (reused by layer 2)
  float* asrc1 = ws + o; o += (size_t)N * 4;
  float* adst1 = ws + o; o += (size_t)N * 4;
  float* m1    = ws + o; o += (size_t)N * 4;
  float* den1  = ws + o; o += (size_t)N * 4;
  float* asrc2 = ws + o; o += (size_t)N;
  float* adst2 = ws + o; o += (size_t)N;
  float* m2    = ws + o; o += (size_t)N;
  float* den2  = ws + o; o += (size_t)N;
  float* g     = ws + o; o += 32;
  float* h2    = h1;                               // reuse: h1 dead after layer-1 agg
  float* out2  = h1 + (size_t)N * 32;

  const int B256 = 256;

  // ================= Layer 1 =================
  {
    long tiles = (long)(N / 16) * (128 / 16);
    wmma_gemm_f32<<<(unsigned)cdivl(tiles * 32, B256), B256, 0, stream>>>(x, W1, h1, N, FIN, 128);
  }
  attn_coef<<<(unsigned)cdivl((long)N * NHEADS, B256), B256, 0, stream>>>(
      h1, att_src1, att_dst1, asrc1, adst1, N, NHEADS, HIDC);
  fill_f32<<<(unsigned)cdivl((long)N * 4, B256), B256, 0, stream>>>(m1, -INFINITY, (long)N * 4);
  fill_f32<<<(unsigned)cdivl((long)N * 4, B256), B256, 0, stream>>>(den1, 0.f, (long)N * 4);
  fill_f32<<<(unsigned)cdivl((long)N * 128, B256), B256, 0, stream>>>(out1, 0.f, (long)N * 128);

  long eh1 = (long)EA * NHEADS;
  edge_score_max<<<(unsigned)cdivl(eh1, B256), B256, 0, stream>>>(
      srcI, dstI, E, N, NHEADS, asrc1, adst1, alpha, m1);
  edge_exp_sum<<<(unsigned)cdivl(eh1, B256), B256, 0, stream>>>(
      srcI, dstI, E, N, NHEADS, alpha, m1, den1);
  edge_norm<<<(unsigned)cdivl(eh1, B256), B256, 0, stream>>>(
      srcI, dstI, E, N, NHEADS, alpha, den1);
  edge_agg<<<(unsigned)cdivl((long)EA * 128, B256), B256, 0, stream>>>(
      srcI, dstI, E, N, NHEADS, HIDC, h1, alpha, out1);
  bias_elu<<<(unsigned)cdivl((long)N * 128, B256), B256, 0, stream>>>(out1, b1, N, 128);

  // ================= Layer 2 =================
  {
    long tiles = (long)(N / 16) * (32 / 16);
    wmma_gemm_f32<<<(unsigned)cdivl(tiles * 32, B256), B256, 0, stream>>>(out1, W2, h2, N, 128, 32);
  }
  attn_coef<<<(unsigned)cdivl((long)N, B256), B256, 0, stream>>>(
      h2, att_src2, att_dst2, asrc2, adst2, N, 1, HIDC);
  fill_f32<<<(unsigned)cdivl((long)N, B256), B256, 0, stream>>>(m2, -INFINITY, (long)N);
  fill_f32<<<(unsigned)cdivl((long)N, B256), B256, 0, stream>>>(den2, 0.f, (long)N);
  fill_f32<<<(unsigned)cdivl((long)N * 32, B256), B256, 0, stream>>>(out2, 0.f, (long)N * 32);
  fill_f32<<<1, 32, 0, stream>>>(g, 0.f, 32);

  long eh2 = (long)EA;
  edge_score_max<<<(unsigned)cdivl(eh2, B256), B256, 0, stream>>>(
      srcI, dstI, E, N, 1, asrc2, adst2, alpha, m2);
  edge_exp_sum<<<(unsigned)cdivl(eh2, B256), B256, 0, stream>>>(
      srcI, dstI, E, N, 1, alpha, m2, den2);
  edge_norm<<<(unsigned)cdivl(eh2, B256), B256, 0, stream>>>(
      srcI, dstI, E, N, 1, alpha, den2);
  edge_agg<<<(unsigned)cdivl((long)EA * 32, B256), B256, 0, stream>>>(
      srcI, dstI, E, N, 1, HIDC, h2, alpha, out2);

  // ================= Readout =================
  colsum32<<<512, B256, 0, stream>>>(out2, g, N);
  finalize<<<1, 1, 0, stream>>>(g, b2, linW, linb, (float*)d_out, N);
}